// Attention_81836306858296
// MI455X (gfx1250) — compile-verified
//
#include <hip/hip_runtime.h>

#define B_ 32
#define C_ 384
#define H_ 64
#define W_ 64
#define L_ 128
#define R_ 24

typedef float v2f __attribute__((ext_vector_type(2)));
typedef float v4f __attribute__((ext_vector_type(4)));
typedef float v8f __attribute__((ext_vector_type(8)));

// 1/sqrt(1 + 1e-5)  (BatchNorm eval scale with mean=0, var=1)
#define BN_RS 0.9999950000374997f

// ---------------------------------------------------------------------------
// Kernel 1: coordinate pooling. One block per (b,c) plane of 64x64.
// y0[b,c,0:64]  = mean over w (row means)
// y0[b,c,64:128]= mean over h (col means)
// LDS tile padded 64x65 -> conflict-free row AND column reads.
// x loads use default (RT) policy: we WANT x resident in L2 (192MB) so the
// final gate pass re-reads it from L2 instead of HBM.
// ---------------------------------------------------------------------------
__global__ __launch_bounds__(256) void pool_kernel(const float* __restrict__ x,
                                                   float* __restrict__ y0) {
  __shared__ float tile[64 * 65];
  const int plane = blockIdx.x;                // b*C + c
  const size_t base = (size_t)plane * 4096;
  for (int i = threadIdx.x; i < 4096; i += 256) {
    const int h = i >> 6, w = i & 63;
    tile[h * 65 + w] = x[base + i];
  }
  __syncthreads();
  const int t = threadIdx.x;
  if (t < 64) {                                // mean over width
    float s = 0.f;
#pragma unroll 8
    for (int w = 0; w < 64; ++w) s += tile[t * 65 + w];
    y0[(size_t)plane * 128 + t] = s * (1.0f / 64.0f);
  } else if (t < 128) {                        // mean over height
    const int w = t - 64;
    float s = 0.f;
#pragma unroll 8
    for (int h = 0; h < 64; ++h) s += tile[h * 65 + w];
    y0[(size_t)plane * 128 + 64 + w] = s * (1.0f / 64.0f);
  }
}

// ---------------------------------------------------------------------------
// Kernel 2: generic 1x1-conv + BN(eval) + ReLU as an f32 WMMA GEMM.
//   out[b, m, lcol] = relu( (sum_k Wm[m,k] * in[b, k, in_col_off+lcol]) * g[m]*rs + bias[m] )
// One wave per 16x16 output tile, K-loop of V_WMMA_F32_16X16X4_F32.
// Fragment layouts per CDNA5 ISA 7.12.2:
//   A 16x4 f32 : lanes 0-15 -> M=lane, v0=K0,v1=K1 ; lanes 16-31 -> v0=K2,v1=K3
//   B 4x16 f32 : lanes 0-15 -> N=lane, v0=K0,v1=K1 ; lanes 16-31 -> v0=K2,v1=K3
//   C/D 16x16  : VGPR r, lanes 0-15 -> (M=r, N=lane); lanes 16-31 -> (M=r+8)
// ---------------------------------------------------------------------------
__global__ __launch_bounds__(256) void cbr_wmma_kernel(
    const float* __restrict__ Wm, const float* __restrict__ g,
    const float* __restrict__ bias, const float* __restrict__ in,
    float* __restrict__ out, int ncols_pb, int in_col_off, int n_tiles_m,
    int n_tiles_n) {
  const int wave = (int)((blockIdx.x * blockDim.x + threadIdx.x) >> 5);
  const int lane = threadIdx.x & 31;
  if (wave >= n_tiles_m * n_tiles_n) return;   // uniform per wave
  const int mt = wave % n_tiles_m;
  const int nt = wave / n_tiles_m;
  const int half = lane >> 4;
  const int l16 = lane & 15;

  const int mrow = mt * 16 + l16;              // A-matrix row for this lane
  const int ncol = nt * 16 + l16;              // global output column
  const int b = ncol / ncols_pb;
  const int lcol = ncol - b * ncols_pb;
  // input is always [B, C, 128] with row stride 128
  const size_t in_colbase = (size_t)b * (C_ * 128) + (size_t)(in_col_off + lcol);
  const size_t wrow = (size_t)mrow * C_;

  v8f acc = {0.f, 0.f, 0.f, 0.f, 0.f, 0.f, 0.f, 0.f};
#pragma unroll 4
  for (int k0 = 0; k0 < C_; k0 += 4) {
    const int kA = k0 + half * 2;
    v2f a, bb;
    a.x = Wm[wrow + kA];
    a.y = Wm[wrow + kA + 1];
    bb.x = in[in_colbase + (size_t)kA * 128];
    bb.y = in[in_colbase + (size_t)(kA + 1) * 128];
    acc = __builtin_amdgcn_wmma_f32_16x16x4_f32(false, a, false, bb, (short)0,
                                                acc, false, false);
  }

  const size_t out_b = (size_t)b * C_ * ncols_pb;
#pragma unroll
  for (int r = 0; r < 8; ++r) {
    const int m = mt * 16 + half * 8 + r;
    float v = acc[r] * (g[m] * BN_RS) + bias[m];
    v = v > 0.f ? v : 0.f;
    out[out_b + (size_t)m * ncols_pb + lcol] = v;
  }
}

// ---------------------------------------------------------------------------
// Kernel 3: wavelet (scaled channel means) + 384->24->384 MLP. One block per b.
// z[b,c] = 2^((chunk+1)/2) * mean_l y[b,c,l], chunk = c/128
// ---------------------------------------------------------------------------
__global__ __launch_bounds__(384) void mlp_kernel(
    const float* __restrict__ y, const float* __restrict__ fc0_w,
    const float* __restrict__ fc0_b, const float* __restrict__ bn1_g,
    const float* __restrict__ bn1_b, const float* __restrict__ fc1_w,
    const float* __restrict__ fc1_b, float* __restrict__ z2) {
  __shared__ float zvec[C_];
  __shared__ float z1[R_];
  const int b = blockIdx.x, c = threadIdx.x;
  const size_t yb = (size_t)b * C_ * 128 + (size_t)c * 128;
  float s = 0.f;
#pragma unroll 8
  for (int l = 0; l < 128; ++l) s += y[yb + l];
  const int chunk = c >> 7;
  zvec[c] = s * exp2f(0.5f * (float)(chunk + 1)) * (1.0f / 128.0f);
  __syncthreads();
  if (c < R_) {
    float d = fc0_b[c];
    for (int k = 0; k < C_; ++k) d += fc0_w[c * C_ + k] * zvec[k];
    d = d * (bn1_g[c] * BN_RS) + bn1_b[c];
    z1[c] = d > 0.f ? d : 0.f;
  }
  __syncthreads();
  float o = fc1_b[c];
#pragma unroll
  for (int r = 0; r < R_; ++r) o += fc1_w[c * R_ + r] * z1[r];
  z2[(size_t)b * C_ + c] = o;
}

// ---------------------------------------------------------------------------
// Kernel 4: out = x * (sigmoid(xh[b,c,h] * xw[b,c,w]) + z2[b,c])
// float4-vectorized bandwidth-dominant pass.
//  - x is read once more and never again -> non-temporal load (last use),
//    served mostly from L2 (left resident by the pool pass).
//  - out is streamed once and never re-read -> non-temporal store so the
//    201MB of store traffic doesn't evict still-unread x lines from L2.
// ---------------------------------------------------------------------------
__global__ __launch_bounds__(256) void final_kernel(
    const float* __restrict__ x, const float* __restrict__ xh,
    const float* __restrict__ xw, const float* __restrict__ z2,
    float* __restrict__ out) {
  const size_t idx = (size_t)blockIdx.x * blockDim.x + threadIdx.x;
  size_t t = idx;
  const int w4 = (int)(t & 15); t >>= 4;      // 4 consecutive w per thread
  const int h = (int)(t & 63);  t >>= 6;
  const int plane = (int)t;                   // b*C + c
  const v4f xv = __builtin_nontemporal_load(
      reinterpret_cast<const v4f*>(x) + idx);
  const float hv = xh[(size_t)plane * 64 + h];
  const v4f wv = reinterpret_cast<const v4f*>(xw)[(size_t)plane * 16 + w4];
  const float zz = z2[plane];
  v4f o;
  o.x = xv.x * (1.f / (1.f + __expf(-hv * wv.x)) + zz);
  o.y = xv.y * (1.f / (1.f + __expf(-hv * wv.y)) + zz);
  o.z = xv.z * (1.f / (1.f + __expf(-hv * wv.z)) + zz);
  o.w = xv.w * (1.f / (1.f + __expf(-hv * wv.w)) + zz);
  __builtin_nontemporal_store(o, reinterpret_cast<v4f*>(out) + idx);
}

// ---------------------------------------------------------------------------
extern "C" void kernel_launch(void* const* d_in, const int* in_sizes, int n_in,
                              void* d_out, int out_size, void* d_ws,
                              size_t ws_size, hipStream_t stream) {
  const float* x     = (const float*)d_in[0];
  const float* Wy    = (const float*)d_in[1];
  const float* gy    = (const float*)d_in[2];
  const float* by    = (const float*)d_in[3];
  const float* Wh    = (const float*)d_in[4];
  const float* gh    = (const float*)d_in[5];
  const float* bh    = (const float*)d_in[6];
  const float* Ww    = (const float*)d_in[7];
  const float* gw    = (const float*)d_in[8];
  const float* bw    = (const float*)d_in[9];
  const float* fc0_w = (const float*)d_in[10];
  const float* fc0_b = (const float*)d_in[11];
  const float* bn1_g = (const float*)d_in[12];
  const float* bn1_b = (const float*)d_in[13];
  const float* fc1_w = (const float*)d_in[14];
  const float* fc1_b = (const float*)d_in[15];
  float* outp = (float*)d_out;

  float* ws = (float*)d_ws;
  float* y0 = ws;                              // [B,C,128] = 1,572,864 f
  float* y  = y0 + (size_t)B_ * C_ * 128;      // [B,C,128]
  float* xh = y  + (size_t)B_ * C_ * 128;      // [B,C,64]  =   786,432 f
  float* xw = xh + (size_t)B_ * C_ * 64;       // [B,C,64]
  float* z2 = xw + (size_t)B_ * C_ * 64;       // [B,C]     =    12,288 f

  // 1) coordinate pooling: one block per (b,c) plane
  pool_kernel<<<B_ * C_, 256, 0, stream>>>(x, y0);

  // 2) y = CBR(y0, Wy): M=384, K=384, N = 32*128 = 4096 -> 24x256 tiles
  cbr_wmma_kernel<<<(24 * 256) / 8, 256, 0, stream>>>(Wy, gy, by, y0, y,
                                                      128, 0, 24, 256);
  // 3) xh = CBR(y[:, :, :64], Wh): N = 32*64 = 2048 -> 24x128 tiles
  cbr_wmma_kernel<<<(24 * 128) / 8, 256, 0, stream>>>(Wh, gh, bh, y, xh,
                                                      64, 0, 24, 128);
  // 4) xw = CBR(y[:, :, 64:], Ww)
  cbr_wmma_kernel<<<(24 * 128) / 8, 256, 0, stream>>>(Ww, gw, bw, y, xw,
                                                      64, 64, 24, 128);
  // 5) wavelet + squeeze-excite MLP
  mlp_kernel<<<B_, 384, 0, stream>>>(y, fc0_w, fc0_b, bn1_g, bn1_b, fc1_w,
                                     fc1_b, z2);
  // 6) gated output (bandwidth-bound pass, NT load of x / NT store of out)
  const int n4 = (B_ * C_ * H_ * W_) / 4;      // 12,582,912
  final_kernel<<<n4 / 256, 256, 0, stream>>>(x, xh, xw, z2, outp);
}